// GFCN_17867063951709
// MI455X (gfx1250) — compile-verified
//
#include <hip/hip_runtime.h>
#include <hip/hip_bf16.h>
#include <math.h>

typedef __attribute__((ext_vector_type(2))) float v2f;
typedef __attribute__((ext_vector_type(8))) float v8f;

static inline int nblk(long long n, int t) { return (int)((n + t - 1) / t); }

// ---------------------------------------------------------------------------
// Generic fill
// ---------------------------------------------------------------------------
__global__ void k_fill(float* __restrict__ p, float v, int n) {
    int i = blockIdx.x * blockDim.x + threadIdx.x;
    if (i < n) p[i] = v;
}

// ---------------------------------------------------------------------------
// Cartesian edge attributes: pseudo = (pos[src]-pos[dst]) / (2*max|.|) + 0.5
// ---------------------------------------------------------------------------
__global__ void k_edge_absmax(const int* __restrict__ ei, const float* __restrict__ pos,
                              int E, float* __restrict__ red) {
    int e = blockIdx.x * blockDim.x + threadIdx.x;
    if (e >= E) return;
    int s = ei[e], d = ei[E + e];
    float dx = fabsf(pos[2 * s]     - pos[2 * d]);
    float dy = fabsf(pos[2 * s + 1] - pos[2 * d + 1]);
    float m = fmaxf(dx, dy);
    atomicMax((int*)red, __float_as_int(m));   // valid for non-negative floats
}

__global__ void k_edge_cart(const int* __restrict__ ei, const float* __restrict__ pos,
                            int E, const float* __restrict__ red, float* __restrict__ ea) {
    int e = blockIdx.x * blockDim.x + threadIdx.x;
    if (e >= E) return;
    float inv = 0.5f / red[0];
    int s = ei[e], d = ei[E + e];
    ea[2 * e]     = (pos[2 * s]     - pos[2 * d])     * inv + 0.5f;
    ea[2 * e + 1] = (pos[2 * s + 1] - pos[2 * d + 1]) * inv + 0.5f;
}

// In-degree per node (count of edges whose dst == n)
__global__ void k_count_deg(const int* __restrict__ ei, int E, float* __restrict__ deg) {
    int e = blockIdx.x * blockDim.x + threadIdx.x;
    if (e >= E) return;
    atomicAdd(&deg[ei[E + e]], 1.0f);
}

// ---------------------------------------------------------------------------
// Repack spline weights + root into a zero-padded row-major [Kp, Npad] GEMM
// operand (Kp = Ci rounded to 4, Npad = (ks2+1)*Co rounded to 64):
//   Wg[ci, k*Co+co] = W[k, ci, co]   (k < ks2)
//   Wg[ci, ks2*Co+co] = root[ci, co]
//   everything else = 0
// ---------------------------------------------------------------------------
__global__ void k_repack(const float* __restrict__ W, const float* __restrict__ root,
                         int ks2, int Ci, int Co, int Kp, int Npad,
                         float* __restrict__ Wg) {
    int i = blockIdx.x * blockDim.x + threadIdx.x;
    if (i >= Kp * Npad) return;
    int nn = i % Npad;
    int ci = i / Npad;
    int Nt = (ks2 + 1) * Co;
    float v = 0.0f;
    if (ci < Ci && nn < Nt) {
        int co = nn % Co;
        int k  = nn / Co;
        v = (k < ks2) ? W[((size_t)k * Ci + ci) * Co + co] : root[(size_t)ci * Co + co];
    }
    Wg[i] = v;
}

// ---------------------------------------------------------------------------
// f32 GEMM C[M,N] = A[M,K] @ B[Kp,Npad] via V_WMMA_F32_16X16X4_F32 (wave32).
// B is zero-padded so all panel loads are unguarded. One block = 4 waves,
// computing a 16 x 64 output tile. A tile (16 x Kp) and B panel (Kp x 64)
// are staged in LDS; B stored k-pair interleaved so each lane's (ka, ka+1)
// pair is one ds_load_b64. Inner loop is branchless: 2x ds_load_b64 + wmma.
// ---------------------------------------------------------------------------
__global__ void k_gemm(const float* __restrict__ A, const float* __restrict__ B,
                       float* __restrict__ C, int M, int N, int Npad, int K) {
    __shared__ float As[16][128];
    __shared__ float Bs[8192];   // [(k/2)*128 + n*2 + (k&1)], k < Kp <= 128, n < 64
    const int tileM = blockIdx.y;
    const int n0    = blockIdx.x * 64;
    const int Kp    = (K + 3) & ~3;

    // stage A tile (guards only matter at the M / K boundary)
    for (int i = threadIdx.x; i < 16 * Kp; i += 128) {
        int r = i / Kp, k = i % Kp;
        int m = tileM * 16 + r;
        As[r][k] = (m < M && k < K) ? A[(size_t)m * K + k] : 0.0f;
    }
    // stage B panel, unguarded (B padded to Kp x Npad, n0+63 < Npad)
    for (int i = threadIdx.x; i < Kp * 64; i += 128) {
        int k = i / 64, nn = i % 64;
        Bs[(k >> 1) * 128 + nn * 2 + (k & 1)] = B[(size_t)k * Npad + n0 + nn];
    }
    __syncthreads();

    const int wave = threadIdx.x >> 5;
    const int lane = threadIdx.x & 31;
    const int half = lane >> 4;
    const int l16  = lane & 15;
    const int nLoc = wave * 16 + l16;
    const int n    = n0 + nLoc;

    v8f acc = {};
#pragma unroll 2
    for (int k0 = 0; k0 < Kp; k0 += 4) {
        int ka = k0 + 2 * half;                       // even
        v2f a = *(const v2f*)&As[l16][ka];            // ds_load_b64
        v2f b = *(const v2f*)&Bs[(ka >> 1) * 128 + nLoc * 2];  // ds_load_b64
        acc = __builtin_amdgcn_wmma_f32_16x16x4_f32(
            /*neg_a=*/false, a, /*neg_b=*/false, b,
            /*c_mod=*/(short)0, acc, /*reuse_a=*/false, /*reuse_b=*/false);
    }

    if (n < N) {
#pragma unroll
        for (int r = 0; r < 8; ++r) {
            int m = tileM * 16 + r + half * 8;
            if (m < M) C[(size_t)m * N + n] = acc[r];
        }
    }
}

// ---------------------------------------------------------------------------
// Bilinear spline message scatter (ks > 1): for each (edge, channel),
// combine 4 basis slices of Y[src] and atomically add into agg[dst].
// ---------------------------------------------------------------------------
__global__ void k_scatter(const int* __restrict__ ei, const float* __restrict__ ea,
                          const float* __restrict__ Y, int E, int ks, int Co, int ldY,
                          float* __restrict__ agg) {
    int t = blockIdx.x * blockDim.x + threadIdx.x;
    if (t >= E * Co) return;
    int e = t / Co, c = t % Co;
    int s = ei[e], d = ei[E + e];
    float vx = ea[2 * e]     * (ks - 1.0f);
    float vy = ea[2 * e + 1] * (ks - 1.0f);
    int lx = (int)floorf(vx); lx = lx < 0 ? 0 : (lx > ks - 2 ? ks - 2 : lx);
    int ly = (int)floorf(vy); ly = ly < 0 ? 0 : (ly > ks - 2 ? ks - 2 : ly);
    float fx = vx - (float)lx, fy = vy - (float)ly;
    const float* y = Y + (size_t)s * ldY;
    float m = (1.0f - fx) * (1.0f - fy) * y[((lx    ) * ks + ly    ) * Co + c]
            + (1.0f - fx) * fy          * y[((lx    ) * ks + ly + 1) * Co + c]
            + fx          * (1.0f - fy) * y[((lx + 1) * ks + ly    ) * Co + c]
            + fx          * fy          * y[((lx + 1) * ks + ly + 1) * Co + c];
    atomicAdd(&agg[(size_t)d * Co + c], m);
}

// ks == 1: message is simply Y[src, 0:Co]
__global__ void k_gather1(const int* __restrict__ ei, const float* __restrict__ Y,
                          int E, int Co, int ldY, float* __restrict__ agg) {
    int t = blockIdx.x * blockDim.x + threadIdx.x;
    if (t >= E * Co) return;
    int e = t / Co, c = t % Co;
    int s = ei[e], d = ei[E + e];
    atomicAdd(&agg[(size_t)d * Co + c], Y[(size_t)s * ldY + c]);
}

// out = agg/max(deg,1) + rootTerm + bias  [optional ELU, optional accumulate]
__global__ void k_finalize(const float* __restrict__ agg, const float* __restrict__ Y,
                           const float* __restrict__ deg, const float* __restrict__ bias,
                           int n, int Co, int ldY, int rootoff, int do_elu, int accum,
                           float* __restrict__ out) {
    int t = blockIdx.x * blockDim.x + threadIdx.x;
    if (t >= n * Co) return;
    int i = t / Co, c = t % Co;
    float v = agg[t] / fmaxf(deg[i], 1.0f) + Y[(size_t)i * ldY + rootoff + c] + bias[c];
    if (do_elu) v = v > 0.0f ? v : (expf(v) - 1.0f);
    out[t] = accum ? (out[t] + v) : v;
}

// ---------------------------------------------------------------------------
// Segment ops
// ---------------------------------------------------------------------------
__global__ void k_segacc(const float* __restrict__ xin, const int* __restrict__ cl,
                         int n, int C, float* __restrict__ sum, float* __restrict__ cnt) {
    int t = blockIdx.x * blockDim.x + threadIdx.x;
    if (t >= n * C) return;
    int i = t / C, c = t % C;
    atomicAdd(&sum[(size_t)cl[i] * C + c], xin[t]);
    if (c == 0) atomicAdd(&cnt[cl[i]], 1.0f);
}

__global__ void k_segdiv(float* __restrict__ sum, const float* __restrict__ cnt, int nc, int C) {
    int t = blockIdx.x * blockDim.x + threadIdx.x;
    if (t >= nc * C) return;
    sum[t] = sum[t] / fmaxf(cnt[t / C], 1.0f);
}

// w = cnt[cl] * x / (g[cl] + 0.001), NaN -> 0
__global__ void k_pw(const float* __restrict__ xin, const int* __restrict__ cl,
                     const float* __restrict__ g, const float* __restrict__ cnt,
                     int n, int C, float* __restrict__ w) {
    int t = blockIdx.x * blockDim.x + threadIdx.x;
    if (t >= n * C) return;
    int i = t / C, c = t % C;
    float v = cnt[cl[i]] * xin[t] / (g[(size_t)cl[i] * C + c] + 0.001f);
    w[t] = (v != v) ? 0.0f : v;
}

// u[n,c] = src[cl[n], c] * w[n, (wC==1 ? 0 : c)]
__global__ void k_unpool(const float* __restrict__ src, const int* __restrict__ cl,
                         const float* __restrict__ w, int n, int C, int wC,
                         float* __restrict__ out) {
    int t = blockIdx.x * blockDim.x + threadIdx.x;
    if (t >= n * C) return;
    int i = t / C, c = t % C;
    out[t] = src[(size_t)cl[i] * C + c] * w[(size_t)i * wC + (wC == 1 ? 0 : c)];
}

// ---------------------------------------------------------------------------
// BatchNorm1d (training mode: batch mean, biased var). One block per channel.
// ---------------------------------------------------------------------------
__global__ void k_bn(float* __restrict__ xio, const float* __restrict__ gamma,
                     const float* __restrict__ beta, int n, int C) {
    __shared__ float sb[256];
    int c = blockIdx.x;
    float s = 0.0f;
    for (int i = threadIdx.x; i < n; i += 256) s += xio[(size_t)i * C + c];
    sb[threadIdx.x] = s;
    __syncthreads();
    for (int o = 128; o > 0; o >>= 1) {
        if (threadIdx.x < o) sb[threadIdx.x] += sb[threadIdx.x + o];
        __syncthreads();
    }
    float mean = sb[0] / (float)n;
    __syncthreads();
    float vs = 0.0f;
    for (int i = threadIdx.x; i < n; i += 256) {
        float d = xio[(size_t)i * C + c] - mean;
        vs += d * d;
    }
    sb[threadIdx.x] = vs;
    __syncthreads();
    for (int o = 128; o > 0; o >>= 1) {
        if (threadIdx.x < o) sb[threadIdx.x] += sb[threadIdx.x + o];
        __syncthreads();
    }
    float inv = rsqrtf(sb[0] / (float)n + 1e-5f);
    float ga = gamma[c], be = beta[c];
    for (int i = threadIdx.x; i < n; i += 256)
        xio[(size_t)i * C + c] = (xio[(size_t)i * C + c] - mean) * inv * ga + be;
}

// ===========================================================================
// Host orchestration
// ===========================================================================
extern "C" void kernel_launch(void* const* d_in, const int* in_sizes, int n_in,
                              void* d_out, int out_size, void* d_ws, size_t ws_size,
                              hipStream_t stream) {
    struct Conv { const float *W, *root, *bias; };
    auto conv_at = [&](int i) {
        return Conv{(const float*)d_in[i], (const float*)d_in[i + 1], (const float*)d_in[i + 2]};
    };

    const float* x = (const float*)d_in[0];
    Conv c1a = conv_at(1),  c1b = conv_at(4);
    Conv c2a = conv_at(7),  c2b = conv_at(10);
    const float* bn2g = (const float*)d_in[13]; const float* bn2b = (const float*)d_in[14];
    Conv c3a = conv_at(15), c3b = conv_at(18);
    const float* bn3g = (const float*)d_in[21]; const float* bn3b = (const float*)d_in[22];
    Conv fr1 = conv_at(23), fr2 = conv_at(26), sp2 = conv_at(29), cvo = conv_at(32);

    const float* pos[4]; const int* ei[4]; const int* cl[3];
    int n[4], E[4];
    for (int l = 0; l < 4; ++l) {
        pos[l] = (const float*)d_in[35 + l];
        ei[l]  = (const int*)d_in[39 + l];
        n[l] = in_sizes[35 + l] / 2;
        E[l] = in_sizes[39 + l] / 2;
    }
    for (int l = 0; l < 3; ++l) cl[l] = (const int*)d_in[43 + l];
    int n1 = n[0], n2 = n[1], n3 = n[2], n4 = n[3];

    // ---- bump allocator over d_ws ----
    char* base = (char*)d_ws;
    size_t off = 0;
    auto alloc = [&](size_t nf) -> float* {
        float* p = (float*)(base + off);
        off += ((nf * sizeof(float) + 255) & ~(size_t)255);
        return p;
    };
    auto imax = [](size_t a, size_t b) { return a > b ? a : b; };

    float* ea[4]; float* deg[4];
    for (int l = 0; l < 4; ++l) { ea[l] = alloc((size_t)2 * E[l]); deg[l] = alloc(n[l]); }
    float* red  = alloc(4);
    float* Wg   = alloc((size_t)128 * 896);          // >= largest Kp*Npad used (64 x 1280)
    size_t ysz  = imax(imax((size_t)n1 * 832, (size_t)n2 * 640), (size_t)n3 * 1280);
    float* Y    = alloc(ysz);
    size_t asz  = imax(imax((size_t)n1 * 32, (size_t)n2 * 64),
                       imax((size_t)n3 * 128, (size_t)n4 * 64));
    float* agg  = alloc(asz);
    size_t gsz  = imax((size_t)n2, imax((size_t)n3 * 32, (size_t)n4 * 64));
    float* gbuf = alloc(gsz);
    float* cnt  = alloc(imax((size_t)n2, imax((size_t)n3, (size_t)n4)));
    float* hA   = alloc((size_t)n1 * 32);
    float* hB   = alloc((size_t)n1 * 32);
    float* x2   = alloc((size_t)n2 * 32);
    float* h2a  = alloc((size_t)n2 * 64);
    float* h2b  = alloc((size_t)n2 * 64);
    float* x3   = alloc((size_t)n3 * 64);
    float* h3a  = alloc((size_t)n3 * 128);
    float* h3b  = alloc((size_t)n3 * 128);
    float* x4   = alloc((size_t)n4 * 128);
    float* h4   = alloc((size_t)n4 * 64);
    float* u3   = alloc((size_t)n3 * 64);
    float* u3b  = alloc((size_t)n3 * 32);
    float* u2   = alloc((size_t)n2 * 32);
    float* u1   = alloc((size_t)n1 * 32);
    float* w1   = alloc(n1);
    float* w2   = alloc((size_t)n2 * 32);
    float* w3   = alloc((size_t)n3 * 64);

    const int T = 256;
    auto fill = [&](float* p, float v, long long cntN) {
        k_fill<<<nblk(cntN, T), T, 0, stream>>>(p, v, (int)cntN);
    };

    // ---- edge attrs + degrees per level ----
    for (int l = 0; l < 4; ++l) {
        fill(red, 0.0f, 1);
        k_edge_absmax<<<nblk(E[l], T), T, 0, stream>>>(ei[l], pos[l], E[l], red);
        k_edge_cart<<<nblk(E[l], T), T, 0, stream>>>(ei[l], pos[l], E[l], red, ea[l]);
        fill(deg[l], 0.0f, n[l]);
        k_count_deg<<<nblk(E[l], T), T, 0, stream>>>(ei[l], E[l], deg[l]);
    }

    // ---- spline conv launcher ----
    auto conv = [&](const float* xin, int nn, int Ci, int Co, int ks, Conv P,
                    int lev, int do_elu, int accum, float* outp) {
        int ks2  = ks * ks;
        int Nt   = (ks2 + 1) * Co;
        int Kp   = (Ci + 3) & ~3;
        int Npad = (Nt + 63) & ~63;
        k_repack<<<nblk((long long)Kp * Npad, T), T, 0, stream>>>(
            P.W, P.root, ks2, Ci, Co, Kp, Npad, Wg);
        dim3 grid((Nt + 63) / 64, (nn + 15) / 16);
        k_gemm<<<grid, 128, 0, stream>>>(xin, Wg, Y, nn, Nt, Npad, Ci);
        fill(agg, 0.0f, (long long)nn * Co);
        if (ks == 1)
            k_gather1<<<nblk((long long)E[lev] * Co, T), T, 0, stream>>>(
                ei[lev], Y, E[lev], Co, Nt, agg);
        else
            k_scatter<<<nblk((long long)E[lev] * Co, T), T, 0, stream>>>(
                ei[lev], ea[lev], Y, E[lev], ks, Co, Nt, agg);
        k_finalize<<<nblk((long long)nn * Co, T), T, 0, stream>>>(
            agg, Y, deg[lev], P.bias, nn, Co, Nt, ks2 * Co, do_elu, accum, outp);
    };
    auto pweights = [&](const float* xin, const int* clu, int nn, int nc, int C, float* wout) {
        fill(gbuf, 0.0f, (long long)nc * C);
        fill(cnt, 0.0f, nc);
        k_segacc<<<nblk((long long)nn * C, T), T, 0, stream>>>(xin, clu, nn, C, gbuf, cnt);
        k_pw<<<nblk((long long)nn * C, T), T, 0, stream>>>(xin, clu, gbuf, cnt, nn, C, wout);
    };
    auto pool = [&](const float* xin, const int* clu, int nn, int nc, int C, float* xout) {
        fill(xout, 0.0f, (long long)nc * C);
        fill(cnt, 0.0f, nc);
        k_segacc<<<nblk((long long)nn * C, T), T, 0, stream>>>(xin, clu, nn, C, xout, cnt);
        k_segdiv<<<nblk((long long)nc * C, T), T, 0, stream>>>(xout, cnt, nc, C);
    };

    // ---- level 1 ----
    pweights(x, cl[0], n1, n2, 1, w1);
    conv(x,  n1, 1,  32, 5, c1a, 0, 1, 0, hA);
    conv(hA, n1, 32, 32, 5, c1b, 0, 1, 0, hB);
    pool(hB, cl[0], n1, n2, 32, x2);

    // ---- level 2 ----
    pweights(x2, cl[1], n2, n3, 32, w2);
    conv(x2,  n2, 32, 64, 3, c2a, 1, 1, 0, h2a);
    conv(h2a, n2, 64, 64, 3, c2b, 1, 1, 0, h2b);
    k_bn<<<64, 256, 0, stream>>>(h2b, bn2g, bn2b, n2, 64);
    pool(h2b, cl[1], n2, n3, 64, x3);

    // ---- level 3 ----
    pweights(x3, cl[2], n3, n4, 64, w3);
    conv(x3,  n3, 64,  128, 3, c3a, 2, 1, 0, h3a);
    conv(h3a, n3, 128, 128, 1, c3b, 2, 1, 0, h3b);
    k_bn<<<128, 256, 0, stream>>>(h3b, bn3g, bn3b, n3, 128);
    pool(h3b, cl[2], n3, n4, 128, x4);

    // ---- decoder ----
    conv(x4, n4, 128, 64, 1, fr1, 3, 1, 0, h4);
    k_unpool<<<nblk((long long)n3 * 64, T), T, 0, stream>>>(h4, cl[2], w3, n3, 64, 64, u3);
    conv(u3, n3, 64, 32, 1, fr2, 2, 1, 0, u3b);
    conv(x3, n3, 64, 32, 3, sp2, 2, 1, 1, u3b);   // u3b += elu(score_pool2(x3))
    k_unpool<<<nblk((long long)n2 * 32, T), T, 0, stream>>>(u3b, cl[1], w2, n2, 32, 32, u2);
    k_unpool<<<nblk((long long)n1 * 32, T), T, 0, stream>>>(u2, cl[0], w1, n1, 32, 1, u1);
    conv(u1, n1, 32, 1, 5, cvo, 0, 0, 0, (float*)d_out);
}